// ProposalLayer_69217692942973
// MI455X (gfx1250) — compile-verified
//
#include <hip/hip_runtime.h>
#include <hip/hip_bf16.h>

// ProposalLayer for MI455X (gfx1250).
// Phase 1: decode 76-float reg rows -> 8-float prop rows, staged through LDS
//          with GLOBAL_LOAD_ASYNC_TO_LDS_B128 (ASYNCcnt) for contiguous 16B traffic.
// Phase 2: per-batch descending stable sort of scores via 4-pass LSD radix sort
//          (wave32 ballot multi-split ranking), then gather props in score order.

#ifndef REGC
#define REGC 76
#endif

#define BIG_B 8
#define BIG_N 131072
#define TILE_ROWS 128            // rows per decode block (76 floats * 4B * 128 = 38,912B LDS)
#define SORT_TILE 1024           // elements per radix block
#define BLOCKS_PER_BATCH (BIG_N / SORT_TILE)   // 128

#if defined(__has_builtin)
#  if __has_builtin(__builtin_amdgcn_global_load_async_to_lds_b128)
#    define HAVE_ASYNC_B128 1
#  endif
#  if __has_builtin(__builtin_amdgcn_s_wait_asynccnt)
#    define HAVE_WAIT_ASYNC 1
#  endif
#endif

// 16-byte int vector, matching the builtin's parameter type from the
// clang diagnostic: "__attribute__((__vector_size__(4 * sizeof(int)))) int __device__ *"
typedef __attribute__((__vector_size__(4 * sizeof(int)))) int v4i_t;
typedef __attribute__((address_space(1))) v4i_t g_v4i;   // global
typedef __attribute__((address_space(3))) v4i_t l_v4i;   // LDS

__device__ __forceinline__ void wait_async0() {
#if defined(HAVE_WAIT_ASYNC)
  __builtin_amdgcn_s_wait_asynccnt(0);
#else
  asm volatile("s_wait_asynccnt 0" ::: "memory");
#endif
}

// ---------------------------------------------------------------------------
// Decode: one block = 128 rows. Stage 128x76 floats (= 2432 float4, address
// float offset is exactly 4*k since 76 = 19*16B) into LDS, then lane t decodes
// row t. Stride-76 LDS rows -> gcd(76,64)=4 -> only 2-way bank conflicts.
// ---------------------------------------------------------------------------
__global__ void __launch_bounds__(TILE_ROWS)
decode_kernel(const float* __restrict__ scores,
              const float* __restrict__ reg,
              const float* __restrict__ xyz,
              const float* __restrict__ anchor,
              float* __restrict__ props,
              unsigned* __restrict__ keys,
              unsigned* __restrict__ vals,
              int N)
{
  __shared__ float tile[TILE_ROWS * REGC];
  const int tid = threadIdx.x;
  const long long row0 = (long long)blockIdx.x * TILE_ROWS;
  const float* gbase = reg + row0 * REGC;

#if defined(HAVE_ASYNC_B128)
  #pragma unroll
  for (int it = 0; it < 19; ++it) {
    int k = it * TILE_ROWS + tid;                 // 0..2431 contiguous float4s
    __builtin_amdgcn_global_load_async_to_lds_b128(
        (g_v4i*)(gbase + 4 * k),
        (l_v4i*)&tile[4 * k],
        0, 0);
  }
  wait_async0();
#else
  #pragma unroll
  for (int it = 0; it < 19; ++it) {
    int k = it * TILE_ROWS + tid;
    float4 v = ((const float4*)gbase)[k];
    *((float4*)&tile[4 * k]) = v;
  }
#endif
  __syncthreads();

  const float* r = &tile[tid * REGC];
  const long long grow = row0 + tid;
  const float s  = scores[grow];
  const float cx = xyz[grow * 3 + 0];
  const float cy = xyz[grow * 3 + 1];
  const float cz = xyz[grow * 3 + 2];
  const float a0 = anchor[0], a1 = anchor[1], a2 = anchor[2];

  // first-occurrence argmax over 12-wide bins (matches jnp.argmax)
  int xb = 0; float xm = r[0];
  #pragma unroll
  for (int j = 1; j < 12; ++j) { float v = r[j];      if (v > xm) { xm = v; xb = j; } }
  int zb = 0; float zm = r[12];
  #pragma unroll
  for (int j = 1; j < 12; ++j) { float v = r[12 + j]; if (v > zm) { zm = v; zb = j; } }

  float px = (float)xb * 0.5f + 0.25f - 3.0f + r[24 + xb] * 0.5f;
  float pz = (float)zb * 0.5f + 0.25f - 3.0f + r[36 + zb] * 0.5f;
  float py = cy + r[48];

  int hb = 0; float hm = r[49];
  #pragma unroll
  for (int j = 1; j < 12; ++j) { float v = r[49 + j]; if (v > hm) { hm = v; hb = j; } }
  const float TWO_PI = 6.28318530717958647692f;
  const float PI_F   = 3.14159265358979323846f;
  const float APC    = TWO_PI / 12.0f;
  float ang = (float)hb * APC + r[61 + hb] * (APC * 0.5f);
  float ry  = fmodf(ang, TWO_PI);
  if (ry < 0.0f) ry += TWO_PI;          // python % semantics: result in [0, 2pi)
  if (ry > PI_F) ry -= TWO_PI;

  float h = r[73] * a0 + a0;
  float w = r[74] * a1 + a1;
  float l = r[75] * a2 + a2;

  float4 p0 = make_float4(px + cx, py + h * 0.5f, pz + cz, h);
  float4 p1 = make_float4(w, l, ry, s);
  float4* pp = (float4*)(props + grow * 8);
  pp[0] = p0; pp[1] = p1;

  // key transform: ascending radix order == descending score, LSD is stable
  unsigned u = __float_as_uint(s);
  unsigned a = (u & 0x80000000u) ? ~u : (u | 0x80000000u);  // ascending map
  keys[grow] = ~a;                                          // invert -> descending
  vals[grow] = (unsigned)(grow - (long long)(grow / N) * N);
}

// ---------------------------------------------------------------------------
// Radix pass 1/3: per-block digit histogram (LDS atomics), digit-major layout
// hist[batch][digit][block]
// ---------------------------------------------------------------------------
__global__ void __launch_bounds__(256)
hist_kernel(const unsigned* __restrict__ keys, unsigned* __restrict__ hist,
            int shift, int N)
{
  __shared__ unsigned h[256];
  const int tid = threadIdx.x;
  const int blk = blockIdx.x, batch = blockIdx.y;
  h[tid] = 0u;
  __syncthreads();
  const unsigned* kb = keys + (long long)batch * N + (long long)blk * SORT_TILE;
  #pragma unroll
  for (int j = 0; j < 4; ++j) {
    unsigned d = (kb[j * 256 + tid] >> shift) & 255u;
    atomicAdd(&h[d], 1u);
  }
  __syncthreads();
  hist[((long long)batch * 256 + tid) * BLOCKS_PER_BATCH + blk] = h[tid];
}

// ---------------------------------------------------------------------------
// Radix pass 2/3: per-batch exclusive scan over 256*128 = 32768 counters
// (one 1024-thread block per batch, 32 chunks of Hillis-Steele)
// ---------------------------------------------------------------------------
__global__ void __launch_bounds__(1024)
scan_kernel(unsigned* __restrict__ hist)
{
  __shared__ unsigned tmp[1024];
  const int tid = threadIdx.x;
  unsigned* seg = hist + (long long)blockIdx.x * (256 * BLOCKS_PER_BATCH);
  unsigned carry = 0u;
  for (int c = 0; c < 32; ++c) {
    unsigned v = seg[c * 1024 + tid];
    tmp[tid] = v;
    __syncthreads();
    for (int off = 1; off < 1024; off <<= 1) {
      unsigned x = tmp[tid];
      unsigned y = (tid >= off) ? tmp[tid - off] : 0u;
      __syncthreads();
      tmp[tid] = x + y;
      __syncthreads();
    }
    unsigned incl  = tmp[tid];
    unsigned total = tmp[1023];
    seg[c * 1024 + tid] = incl - v + carry;   // exclusive + running carry
    __syncthreads();
    carry += total;
  }
}

// ---------------------------------------------------------------------------
// Radix pass 3/3: stable scatter. Wave32 ballot multi-split gives in-wave rank;
// per-wave LDS histograms + serial per-digit wave scan give in-block rank.
// ---------------------------------------------------------------------------
__global__ void __launch_bounds__(1024)
scatter_kernel(const unsigned* __restrict__ ki, const unsigned* __restrict__ vi,
               unsigned* __restrict__ ko, unsigned* __restrict__ vo,
               const unsigned* __restrict__ hist, int shift, int N)
{
  __shared__ unsigned wh[32 * 256];            // 32 KB
  const int tid  = threadIdx.x;
  const int lane = tid & 31, wave = tid >> 5;
  const int blk  = blockIdx.x, batch = blockIdx.y;

  for (int j = tid; j < 32 * 256; j += 1024) wh[j] = 0u;
  __syncthreads();

  const long long base = (long long)batch * N + (long long)blk * SORT_TILE;
  unsigned key = ki[base + tid];
  unsigned val = vi[base + tid];
  unsigned d = (key >> shift) & 255u;

  unsigned mask = 0xFFFFFFFFu;                 // lanes in my wave with my digit
  #pragma unroll
  for (int k = 0; k < 8; ++k) {
    int bit = (d >> k) & 1;
    unsigned b32 = (unsigned)__ballot(bit);    // wave32: low 32 bits
    mask &= bit ? b32 : ~b32;
  }
  unsigned lt   = (1u << lane) - 1u;
  unsigned rank = __popc(mask & lt);
  if (rank == 0) wh[wave * 256 + d] = (unsigned)__popc(mask);
  __syncthreads();

  if (tid < 256) {                             // exclusive scan over waves, per digit
    unsigned run = 0u;
    #pragma unroll
    for (int w = 0; w < 32; ++w) {
      unsigned t = wh[w * 256 + tid];
      wh[w * 256 + tid] = run;
      run += t;
    }
  }
  __syncthreads();

  unsigned posInBlock = wh[wave * 256 + d] + rank;
  unsigned gpos = hist[((long long)batch * 256 + d) * BLOCKS_PER_BATCH + blk] + posInBlock;
  ko[(long long)batch * N + gpos] = key;
  vo[(long long)batch * N + gpos] = val;
}

// ---------------------------------------------------------------------------
// Final gather: out[b][i][:] = props[b][order[b][i]][:]  (two float4 per row)
// ---------------------------------------------------------------------------
__global__ void __launch_bounds__(256)
gather_kernel(const float4* __restrict__ props, const unsigned* __restrict__ ord,
              float4* __restrict__ out, int N)
{
  const long long g = (long long)blockIdx.x * 256 + threadIdx.x;   // row over B*N
  const long long batch = g / N;
  unsigned o = ord[g];
  const float4* src = props + ((long long)batch * N + o) * 2;
  float4* dst = out + g * 2;
  dst[0] = src[0];
  dst[1] = src[1];
}

// ---------------------------------------------------------------------------
extern "C" void kernel_launch(void* const* d_in, const int* in_sizes, int n_in,
                              void* d_out, int out_size, void* d_ws, size_t ws_size,
                              hipStream_t stream) {
  (void)in_sizes; (void)n_in; (void)out_size; (void)ws_size;
  const float* scores = (const float*)d_in[0];   // (B, N) f32
  const float* reg    = (const float*)d_in[1];   // (B, N, 76) f32
  const float* xyz    = (const float*)d_in[2];   // (B, N, 3) f32
  const float* anchor = (const float*)d_in[3];   // (3,) f32

  const int B = BIG_B, N = BIG_N;
  const long long M = (long long)B * N;

  char* ws = (char*)d_ws;
  size_t off = 0;
  float*    props = (float*)(ws + off);    off += (size_t)M * 8 * sizeof(float);
  unsigned* keysA = (unsigned*)(ws + off); off += (size_t)M * sizeof(unsigned);
  unsigned* valsA = (unsigned*)(ws + off); off += (size_t)M * sizeof(unsigned);
  unsigned* keysB = (unsigned*)(ws + off); off += (size_t)M * sizeof(unsigned);
  unsigned* valsB = (unsigned*)(ws + off); off += (size_t)M * sizeof(unsigned);
  unsigned* hist  = (unsigned*)(ws + off); off += (size_t)B * 256 * BLOCKS_PER_BATCH * sizeof(unsigned);

  decode_kernel<<<(int)(M / TILE_ROWS), TILE_ROWS, 0, stream>>>(
      scores, reg, xyz, anchor, props, keysA, valsA, N);

  unsigned *ki = keysA, *vi = valsA, *ko = keysB, *vo = valsB;
  for (int p = 0; p < 4; ++p) {
    const int shift = p * 8;
    hist_kernel<<<dim3(BLOCKS_PER_BATCH, B), 256, 0, stream>>>(ki, hist, shift, N);
    scan_kernel<<<B, 1024, 0, stream>>>(hist);
    scatter_kernel<<<dim3(BLOCKS_PER_BATCH, B), 1024, 0, stream>>>(ki, vi, ko, vo, hist, shift, N);
    unsigned* t;
    t = ki; ki = ko; ko = t;
    t = vi; vi = vo; vo = t;
  }
  // 4 passes -> sorted order lives in valsA (== vi)

  gather_kernel<<<(int)(M / 256), 256, 0, stream>>>(
      (const float4*)props, vi, (float4*)d_out, N);
}